// PRNNLayer_73632919323001
// MI455X (gfx1250) — compile-verified
//
#include <hip/hip_runtime.h>
#include <cstdint>
#include <cstddef>

#ifndef __has_builtin
#define __has_builtin(x) 0
#endif

typedef uint32_t v4u32 __attribute__((ext_vector_type(4)));
typedef int32_t  v8i32 __attribute__((ext_vector_type(8)));
typedef int32_t  v4i32 __attribute__((ext_vector_type(4)));

constexpr int Bn  = 4096;
constexpr int Tn  = 2048;
constexpr int BPB = 32;                 // basins per block = one wave32
constexpr int TC  = 64;                 // timesteps per chunk
constexpr int NCHUNK   = Tn / TC;       // 32
constexpr int IN_ROW   = 3 * TC;        // 192 dwords of input per basin per chunk
constexpr int IN_ROWP  = IN_ROW + IN_ROW / 64;  // 195 (TDM pad: 1 DW per 64 DW)
constexpr int IN_TILE  = BPB * IN_ROWP;         // 6240 dwords per buffer
constexpr int OUT_ROW  = 5 * TC;        // 320
constexpr int OUT_ROWP = OUT_ROW + 1;   // 321 -> conflict-free scalar writes
constexpr int OUT_TILE = BPB * OUT_ROWP;        // 10272
constexpr int SMEM_DW  = 2 * IN_TILE + OUT_TILE;  // 22752 dwords = 91008 bytes

#if defined(__HIP_DEVICE_COMPILE__) && __has_builtin(__builtin_amdgcn_tensor_load_to_lds)
#define PRNN_USE_TDM 1
#else
#define PRNN_USE_TDM 0
#endif

// ---------- fast math helpers (gfx1250 trans ops) ----------
__device__ __forceinline__ float fast_exp2(float x) {
#if defined(__HIP_DEVICE_COMPILE__) && __has_builtin(__builtin_amdgcn_exp2f)
    return __builtin_amdgcn_exp2f(x);
#else
    return exp2f(x);
#endif
}
__device__ __forceinline__ float fast_rcp(float x) {
#if defined(__HIP_DEVICE_COMPILE__) && __has_builtin(__builtin_amdgcn_rcpf)
    return __builtin_amdgcn_rcpf(x);
#else
    return 1.0f / x;
#endif
}
__device__ __forceinline__ float fast_exp(float x) {
    return fast_exp2(x * 1.4426950408889634f);
}
// hv(x) = (tanh(5x)+1)/2 : PROBE -> unconditional v_tanh_f32 on the device pass
__device__ __forceinline__ float hv(float x) {
#if defined(__HIP_DEVICE_COMPILE__)
    return __builtin_amdgcn_tanhf(5.0f * x) * 0.5f + 0.5f;   // v_tanh_f32
#else
    return tanhf(5.0f * x) * 0.5f + 0.5f;                    // host parse only
#endif
}
__device__ __forceinline__ float clipf(float v) {
    return fminf(fmaxf(v, -100000.0f), 100000.0f);
}

// ---------- sync helpers ----------
__device__ __forceinline__ void wait_ds0() {
#if defined(__HIP_DEVICE_COMPILE__)
    asm volatile("s_wait_dscnt 0x0" ::: "memory");
#endif
}
__device__ __forceinline__ void wait_tensorcnt_le(int n) {
#if PRNN_USE_TDM
    if (n == 0) {
#if __has_builtin(__builtin_amdgcn_s_wait_tensorcnt)
        __builtin_amdgcn_s_wait_tensorcnt(0);
#else
        asm volatile("s_wait_tensorcnt 0x0" ::: "memory");
#endif
    } else {
#if __has_builtin(__builtin_amdgcn_s_wait_tensorcnt)
        __builtin_amdgcn_s_wait_tensorcnt(1);
#else
        asm volatile("s_wait_tensorcnt 0x1" ::: "memory");
#endif
    }
    asm volatile("" ::: "memory");
#endif
}

#if PRNN_USE_TDM
// One TDM descriptor: 2-D tile, 192 (dim0, contiguous f32) x 32 (dim1, basins),
// tensor_dim0_stride = T*3 = 6144 elements. data_size=4B.
// pad_enable=1, pad_interval=64 DW (code 5), pad_amount=1 DW (code 0)
// -> LDS rows land at stride 195 dwords: bank-conflict-free lane access.
__device__ __forceinline__ void tdm_load_tile(uint32_t lds_byte_addr, uint64_t gaddr) {
    v4u32 g0;
    g0.x = 1u;                                   // count=1, user mode, no gather
    g0.y = lds_byte_addr;                        // LDS byte address
    g0.z = (uint32_t)(gaddr & 0xFFFFFFFFull);    // global_addr[31:0]
    g0.w = (uint32_t)((gaddr >> 32) & 0x01FFFFFFull) | (2u << 30);  // addr[56:32] | type=2
    // group1: data_size=2(4B) | pad_enable | pad_interval=5 ; dims/strides below
    v8i32 g1 = { (int32_t)0x01520000u,
                 (int32_t)(192u << 16),          // tensor_dim0[15:0] in bits[31:16]
                 (int32_t)(32u  << 16),          // tensor_dim0[31:16]=0 | tensor_dim1[15:0]
                 (int32_t)(192u << 16),          // tensor_dim1[31:16]=0 | tile_dim0=192
                 (int32_t)32,                    // tile_dim1=32 | tile_dim2=0
                 (int32_t)6144,                  // tensor_dim0_stride[31:0]
                 0, 0 };                         // stride0[47:32]=0, stride1=0
    v4i32 z4 = { 0, 0, 0, 0 };
    v8i32 z8 = { 0, 0, 0, 0, 0, 0, 0, 0 };
    asm volatile("" ::: "memory");
    // 6-arg variant: (group0, group1, group2, group3, group4, cpol)
    __builtin_amdgcn_tensor_load_to_lds(g0, g1, z4, z4, z8, 0);
    asm volatile("" ::: "memory");
}
#endif

__global__ __launch_bounds__(BPB, 1) void prnn_scan_kernel(
    const float* __restrict__ in, const float* __restrict__ th, float* __restrict__ out)
{
    extern __shared__ float smem[];
    float* ibuf0 = smem;
    float* ibuf1 = smem + IN_TILE;
    float* obuf  = smem + 2 * IN_TILE;

    const int lane = threadIdx.x;
    const int b0   = blockIdx.x * BPB;

    // theta-derived uniforms (scalar loads)
    const float f_     = th[0]  * 0.1f;
    const float smax_  = th[1]  * 1950.0f;
    const float qmax_  = th[2]  * 50.0f;
    const float ddf_   = th[3]  * 5.0f;
    const float tmin_  = th[4]  * -3.0f;
    const float tmax_  = th[5]  * 3.0f;
    const float Kc_    = th[6]  * 0.5f;
    const float SCmax_ = th[7]  * 1.5f;
    const float spmax_ = th[8]  * 1950.0f;
    const float qpmax_ = th[9]  * 40.0f;
    const float kp     = th[10];
    const float sgmax_ = th[11] * 1950.0f;
    const float qgmax_ = th[12] * 40.0f;
    const float Kl_    = th[13] * 0.5f;
    const float Kn_    = th[14] * 0.5f;
    const float Dc     = 0.986f * 0.862f;
    const float rsmax  = fast_rcp(smax_);

#if PRNN_USE_TDM
    const uint64_t inbase   = (uint64_t)(uintptr_t)in;
    const uint32_t lds_base = (uint32_t)(uintptr_t)(void*)smem;
    // prologue: prefetch chunk 0 into buffer 0
    tdm_load_tile(lds_base, inbase + (uint64_t)b0 * (Tn * 3) * 4ull);
#endif

    float s0v = 0.0f, s1v = 0.0f, s2v = 0.0f, s3v = 0.0f, s4v = 0.0f;

    for (int c = 0; c < NCHUNK; ++c) {
        float* ib = (c & 1) ? ibuf1 : ibuf0;
        const int t0 = c * TC;

#if PRNN_USE_TDM
        if (c + 1 < NCHUNK) {
            wait_ds0();  // drain LDS reads of the buffer the DMA is about to overwrite
            const uint32_t dst = lds_base + (uint32_t)(((c + 1) & 1) ? IN_TILE : 0) * 4u;
            tdm_load_tile(dst, inbase +
                ((uint64_t)b0 * (Tn * 3) + (uint64_t)(c + 1) * IN_ROW) * 4ull);
            wait_tensorcnt_le(1);   // in-order: chunk c's load has landed
        } else {
            wait_tensorcnt_le(0);
        }
#else
        // fallback: cooperative coalesced load into the same padded layout
        for (int r = 0; r < BPB; ++r) {
            const float* src = in + (size_t)(b0 + r) * (Tn * 3) + (size_t)t0 * 3;
            #pragma unroll
            for (int j = 0; j < IN_ROW / 32; ++j) {
                const int x = j * 32 + lane;
                ib[r * IN_ROWP + x + (x >> 6)] = src[x];
            }
        }
        __syncthreads();  // single-wave workgroup: lowers to s_nop
#endif

        const float* myrow = ib + lane * IN_ROWP;
        float* orow = obuf + lane * OUT_ROWP;

        #pragma unroll 4
        for (int i = 0; i < TC; ++i) {
            const int x = 3 * i;
            const float p  = myrow[x     + ( x      >> 6)];
            const float t  = myrow[x + 1 + ((x + 1) >> 6)];
            const float dl = myrow[x + 2 + ((x + 2) >> 6)];

            // PET (Hamon)
            const float pe = (29.8f * 24.0f * 0.611f) * dl *
                             fast_exp(17.3f * t * fast_rcp(t + 237.3f)) *
                             fast_rcp(t + 273.2f);

            // hv(-x) == 1 - hv(x): each opposing pair costs one tanh
            const float day   = hv(dl - 0.5f);
            const float night = 1.0f - day;
            const float scmax_t = SCmax_ * (day * 2.0f + night * 0.6f);
            const float lai_t   = day * 0.478f + night * 0.15f;
            const float kc_t    = Kc_ * (day + night * 0.4f);

            const float hp  = hv(p - 0.1f);
            const float h0  = hv(s0v);
            const float hsc = hv(s0v - scmax_t);
            const float pintc = hp * h0 * hsc * p
                              + hp * (1.0f - hsc) * kc_t * Dc * lai_t;

            const float hsn   = hv(tmin_ - t);
            const float psnow = hsn * p;
            const float prain = (1.0f - hsn) * p;

            const float hmt  = hv(t - tmax_);
            const float h1   = hv(s1v);
            const float melt = hmt * h1 * fminf(s1v, ddf_ * (t - tmax_));

            const float h2  = hv(s2v);
            const float hsp = hv(s2v - spmax_);
            const float qpref = h2 * hsp * qpmax_
                              + h2 * (1.0f - hsp) * kp * p * fast_exp(-f_ * (spmax_ - s2v));

            const float h3  = hv(s3v);
            const float hsm = hv(s3v - smax_);
            const float e3  = fast_exp(-f_ * (smax_ - s3v));
            const float et   = h3 * hsm * pe + h3 * (1.0f - hsm) * pe * (s3v * rsmax);
            const float qsub = h3 * hsm * qmax_ + h3 * (1.0f - hsm) * qmax_ * e3;
            const float qout = h3 * hsm * (s3v - smax_);

            const float h4  = hv(s4v);
            const float hsg = hv(s4v - sgmax_);
            const float qslow = h4 * hsg * qgmax_
                              + h4 * (1.0f - hsg) * (p * Kl_ + p * p * Kn_) *
                                fast_exp(-f_ * (sgmax_ - s4v));

            s0v += clipf(pintc);
            s1v += clipf(psnow - melt);
            s2v += clipf(qpref);
            s3v += clipf(prain + melt - et - qsub - qout);
            s4v += clipf(qslow);

            orow[5 * i + 0] = s0v;
            orow[5 * i + 1] = s1v;
            orow[5 * i + 2] = s2v;
            orow[5 * i + 3] = s3v;
            orow[5 * i + 4] = s4v;
        }

        // coalesced output drain: 128B per store instruction
        for (int r = 0; r < BPB; ++r) {
            float* dst = out + (size_t)(b0 + r) * (Tn * 5) + (size_t)t0 * 5;
            #pragma unroll
            for (int j = 0; j < OUT_ROW / 32; ++j) {
                dst[j * 32 + lane] = obuf[r * OUT_ROWP + j * 32 + lane];
            }
        }
    }
}

extern "C" void kernel_launch(void* const* d_in, const int* in_sizes, int n_in,
                              void* d_out, int out_size, void* d_ws, size_t ws_size,
                              hipStream_t stream) {
    (void)in_sizes; (void)n_in; (void)out_size; (void)d_ws; (void)ws_size;
    const float* in  = (const float*)d_in[0];   // (B, T, 3) f32
    const float* th  = (const float*)d_in[1];   // (15,)     f32
    float* out = (float*)d_out;                 // (B, T, 5) f32
    const size_t shmem = (size_t)SMEM_DW * sizeof(float);   // 91008 B (< 320KB/WGP)
    prnn_scan_kernel<<<dim3(Bn / BPB), dim3(BPB), shmem, stream>>>(in, th, out);
}